// AttentionModuleAggregator_48490180772189
// MI455X (gfx1250) — compile-verified
//
#include <hip/hip_runtime.h>

// Problem constants (reference: K,N,M,E = 32, 8192, 256, 256)
#define K_DIM 32
#define N_DIM 8192
#define M_DIM 256
#define CHUNKS 32                 // N-chunks per k  -> 1024 blocks total
#define CHUNK_N (N_DIM / CHUNKS)  // 256 rows per block

typedef __attribute__((ext_vector_type(2))) float v2f;
typedef __attribute__((ext_vector_type(8))) float v8f;

// ---------------------------------------------------------------------------
// Stage 1: per-(k, n-chunk) partial column sums via V_WMMA_F32_16X16X4_F32
// with an all-ones A matrix:  D[i,j] = sum_p B[p,j] + C[i,j].
// Each wave owns 32 m-columns (two 16-wide WMMA tiles) and sweeps 256 rows,
// 4 rows per WMMA issue. Also writes attention_map = 1.0 (softmax over a
// singleton axis is identically one).
// ---------------------------------------------------------------------------
__global__ __launch_bounds__(256) void attn_stage1_colsum(
    const float* __restrict__ x,   // [K, N, M]
    float* __restrict__ partial,   // [K*CHUNKS, M]
    float* __restrict__ att)       // [K, N]
{
    const int blk   = blockIdx.x;        // 0 .. K*CHUNKS-1
    const int k     = blk / CHUNKS;
    const int chunk = blk % CHUNKS;
    const int tid   = threadIdx.x;       // 256 threads = 8 waves (wave32)
    const int wave  = tid >> 5;
    const int lane  = tid & 31;

    const float* base = x + (size_t)k * ((size_t)N_DIM * M_DIM)
                          + (size_t)chunk * ((size_t)CHUNK_N * M_DIM);

    const int m_base = wave * 32;        // 8 waves * 32 cols = 256 = M
    const int col    = lane & 15;        // column within 16-wide tile
    const int rowsel = (lane >> 4) * 2;  // half-wave 0 -> rows {n,n+1}; half-wave 1 -> {n+2,n+3}

    const float* p = base + (size_t)rowsel * M_DIM + (size_t)m_base + (size_t)col;

    v2f a; a.x = 1.0f; a.y = 1.0f;       // A = ones(16x4): every lane {1,1}
    v8f acc0 = {};
    v8f acc1 = {};

#pragma unroll 4
    for (int i = 0; i < CHUNK_N / 4; ++i) {
        // B tile0: rows n..n+3, cols m_base..m_base+15
        // B tile1: rows n..n+3, cols m_base+16..m_base+31
        v2f b0, b1;
        b0.x = p[0];
        b0.y = p[M_DIM];
        b1.x = p[16];
        b1.y = p[16 + M_DIM];

        acc0 = __builtin_amdgcn_wmma_f32_16x16x4_f32(
            false, a, false, b0, (short)0, acc0, false, false);
        acc1 = __builtin_amdgcn_wmma_f32_16x16x4_f32(
            false, a, false, b1, (short)0, acc1, false, false);

        p += 4 * M_DIM;
    }

    // All 16 rows of D are identical column sums; row M=0 lives in acc[0],
    // lanes 0..15 holding columns 0..15 of the tile.
    if (lane < 16) {
        float* dst = partial + (size_t)blk * M_DIM + m_base;
        dst[col]      = acc0[0];
        dst[16 + col] = acc1[0];
    }

    // attention_map = softmax over a size-1 axis == 1.0 exactly.
    att[(size_t)k * N_DIM + (size_t)chunk * CHUNK_N + tid] = 1.0f;
}

// ---------------------------------------------------------------------------
// Stage 2: fold the CHUNKS partials per (k, m) deterministically.
// ---------------------------------------------------------------------------
__global__ __launch_bounds__(256) void attn_stage2_fold(
    const float* __restrict__ partial,  // [K*CHUNKS, M]
    float* __restrict__ out)            // [K, M]
{
    const int k = blockIdx.x;
    const int m = threadIdx.x;
    const float* p = partial + (size_t)k * CHUNKS * M_DIM + m;
    float s = 0.0f;
#pragma unroll
    for (int c = 0; c < CHUNKS; ++c)
        s += p[(size_t)c * M_DIM];
    out[(size_t)k * M_DIM + m] = s;
}

// ---------------------------------------------------------------------------
// Launch. Inputs in setup_inputs() order: x, W_left, b_left, W_right, b_right,
// w_last, b_last. Only x matters: softmax over the singleton axis makes
// attention_map == 1 and result == sum over N of x. d_out = [K*M result]
// followed by [K*N attention_map].
// ---------------------------------------------------------------------------
extern "C" void kernel_launch(void* const* d_in, const int* in_sizes, int n_in,
                              void* d_out, int out_size, void* d_ws, size_t ws_size,
                              hipStream_t stream) {
    (void)in_sizes; (void)n_in; (void)out_size; (void)ws_size;

    const float* x   = (const float*)d_in[0];
    float* out       = (float*)d_out;                    // result: K*M floats
    float* att       = out + (size_t)K_DIM * M_DIM;      // attention map: K*N floats
    float* partial   = (float*)d_ws;                     // K*CHUNKS*M floats = 1 MB

    attn_stage1_colsum<<<K_DIM * CHUNKS, 256, 0, stream>>>(x, partial, att);
    attn_stage2_fold<<<K_DIM, M_DIM, 0, stream>>>(partial, out);
}